// ModelY_35751307772140
// MI455X (gfx1250) — compile-verified
//
#include <hip/hip_runtime.h>
#include <math.h>

#define N_OBS   6000
#define N_X     6000
#define D_FEAT  16
#define N_DOM   3
#define N_SLICES 16
#define OBS_TILES (N_OBS / 16)   // 375
#define COL_TILES (N_X / 16)     // 375

typedef float v2f __attribute__((ext_vector_type(2)));
typedef float v4f __attribute__((ext_vector_type(4)));
typedef float v8f __attribute__((ext_vector_type(8)));

// ---- workspace layout (float offsets); all offsets are multiples of 16 floats (64B) ----
#define WS_HDR    0                               // [0]=inv_ls, [1..9]=F row-major
#define WS_SXOBS  16                              // xobs / ls          [N_OBS*16]
#define WS_SX     (WS_SXOBS + N_OBS * D_FEAT)     // x / ls             [N_X*16]
#define WS_NOBS   (WS_SX + N_X * D_FEAT)          // ||sxobs_i||^2      [N_OBS]
#define WS_NX     (WS_NOBS + N_OBS)               // ||sx_j||^2         [N_X]
#define WS_BETA0  (WS_NX + N_X)                   // alpha0*F[do][c]    [3*N_OBS]
#define WS_BETA1  (WS_BETA0 + N_DOM * N_OBS)      // alpha1*F[do][c]    [3*N_OBS]
#define WS_PART0  (WS_BETA1 + N_DOM * N_OBS)      // slice partials f0  [16*N_X]
#define WS_PART1  (WS_PART0 + N_SLICES * N_X)     // slice partials f1  [16*N_X]

// ---------------------------------------------------------------------------
// Kernel 1: scalar params -> header. softplus(raw_ls), sigmoid(transfer logits),
// symmetric 3x3 transfer matrix F (triu pair order (0,1),(0,2),(1,2)).
// ---------------------------------------------------------------------------
__global__ void prep_scalars(const float* __restrict__ logits,
                             const float* __restrict__ raw_ls,
                             float* __restrict__ hdr) {
    if (threadIdx.x == 0) {
        float ls = log1pf(__expf(raw_ls[0]));     // jax.nn.softplus
        hdr[0] = 1.0f / ls;
        float t01 = 1.0f / (1.0f + __expf(-logits[0]));
        float t02 = 1.0f / (1.0f + __expf(-logits[1]));
        float t12 = 1.0f / (1.0f + __expf(-logits[2]));
        float F[9] = {1.0f, t01, t02,   t01, 1.0f, t12,   t02, t12, 1.0f};
        #pragma unroll
        for (int k = 0; k < 9; ++k) hdr[1 + k] = F[k];
    }
}

// ---------------------------------------------------------------------------
// Kernel 2: per-row prep. Scale rows by inv_ls, compute row norms, and fold
// alpha * F[dom_o(i), c] into beta arrays (removes all domain logic from the
// hot loop).
// ---------------------------------------------------------------------------
__global__ void prep_rows(const float* __restrict__ xobs,
                          const float* __restrict__ x,
                          const float* __restrict__ alpha0,
                          const float* __restrict__ alpha1,
                          const int*   __restrict__ ro,   // domain_ranges_obs [3][2]
                          float* __restrict__ ws) {
    int tid = blockIdx.x * blockDim.x + threadIdx.x;
    if (tid >= N_OBS + N_X) return;
    const float inv_ls = ws[0];
    if (tid < N_OBS) {
        int i = tid;
        const float* src = xobs + i * D_FEAT;
        float* dst = ws + WS_SXOBS + i * D_FEAT;
        float nrm = 0.0f;
        #pragma unroll
        for (int k = 0; k < D_FEAT; ++k) {
            float v = src[k] * inv_ls;
            dst[k] = v;
            nrm = fmaf(v, v, nrm);
        }
        ws[WS_NOBS + i] = nrm;
        int dom = (i >= ro[2]) + (i >= ro[4]);    // contiguous sorted partition
        float a0 = alpha0[i], a1 = alpha1[i];
        #pragma unroll
        for (int c = 0; c < N_DOM; ++c) {
            float f = ws[1 + dom * 3 + c];
            ws[WS_BETA0 + c * N_OBS + i] = a0 * f;
            ws[WS_BETA1 + c * N_OBS + i] = a1 * f;
        }
    } else {
        int j = tid - N_OBS;
        const float* src = x + j * D_FEAT;
        float* dst = ws + WS_SX + j * D_FEAT;
        float nrm = 0.0f;
        #pragma unroll
        for (int k = 0; k < D_FEAT; ++k) {
            float v = src[k] * inv_ls;
            dst[k] = v;
            nrm = fmaf(v, v, nrm);
        }
        ws[WS_NX + j] = nrm;
    }
}

// ---------------------------------------------------------------------------
// Kernel 3 (hot): fused Matern-5/2 GEMM + contraction.
// One wave = one 16-column x-tile (B operand is loop-invariant, loaded once)
// x one of 16 obs slices. Per 16-obs tile: 4x V_WMMA_F32_16X16X4_F32 build the
// 16x16 Gram tile in fp32; epilogue applies the Matern kernel elementwise and
// accumulates beta0/beta1-weighted column sums. Cross-half reduce via shfl_xor.
// ---------------------------------------------------------------------------
__global__ void __launch_bounds__(256) matern_wmma(const int* __restrict__ rx,  // domain_ranges_x
                                                   float* __restrict__ ws) {
    const int lane = threadIdx.x & 31;
    const int wave = blockIdx.x * (blockDim.x >> 5) + (threadIdx.x >> 5);
    const int colTile = wave >> 4;             // 0..374
    const int slice   = wave & (N_SLICES - 1); // 0..15
    const int n  = lane & 15;                  // N (x column within tile)
    const int g  = lane >> 4;                  // half-wave group
    const int kb = g * 2;                      // K sub-offset for A/B operands
    const int j  = colTile * 16 + n;           // global x column

    const float* __restrict__ sxobs = ws + WS_SXOBS;
    const float* __restrict__ sx    = ws + WS_SX;
    const float* __restrict__ nobs  = ws + WS_NOBS;

    // B operand (4x16 KxN per step, 4 steps over D_FEAT=16) -- loop-invariant.
    const float* bp = sx + j * D_FEAT + kb;
    v2f b0 = *(const v2f*)(bp + 0);
    v2f b1 = *(const v2f*)(bp + 4);
    v2f b2 = *(const v2f*)(bp + 8);
    v2f b3 = *(const v2f*)(bp + 12);
    const float nxl = ws[WS_NX + j];

    const int dx = (j >= rx[2]) + (j >= rx[4]);
    const float* __restrict__ w0p = ws + WS_BETA0 + dx * N_OBS;
    const float* __restrict__ w1p = ws + WS_BETA1 + dx * N_OBS;

    float acc0 = 0.0f, acc1 = 0.0f;

    for (int t = slice; t < OBS_TILES; t += N_SLICES) {
        const int i0 = t * 16;
        // A operand: 16x4 MxK per step; lane holds A[m][kb], A[m][kb+1], m = lane&15
        const float* ap = sxobs + (i0 + n) * D_FEAT + kb;
        v2f a0 = *(const v2f*)(ap + 0);
        v2f a1 = *(const v2f*)(ap + 4);
        v2f a2 = *(const v2f*)(ap + 8);
        v2f a3 = *(const v2f*)(ap + 12);

        v8f c = {};
        c = __builtin_amdgcn_wmma_f32_16x16x4_f32(false, a0, false, b0, (short)0, c, false, false);
        c = __builtin_amdgcn_wmma_f32_16x16x4_f32(false, a1, false, b1, (short)0, c, false, false);
        c = __builtin_amdgcn_wmma_f32_16x16x4_f32(false, a2, false, b2, (short)0, c, false, false);
        c = __builtin_amdgcn_wmma_f32_16x16x4_f32(false, a3, false, b3, (short)0, c, false, false);

        // Row metadata for the 8 M-rows this lane owns (M = r + 8g): 2x b128 each.
        const int rbase = i0 + g * 8;
        v4f nA  = *(const v4f*)(nobs + rbase);
        v4f nB  = *(const v4f*)(nobs + rbase + 4);
        v4f w0A = *(const v4f*)(w0p + rbase);
        v4f w0B = *(const v4f*)(w0p + rbase + 4);
        v4f w1A = *(const v4f*)(w1p + rbase);
        v4f w1B = *(const v4f*)(w1p + rbase + 4);

        float nr[8] = {nA.x, nA.y, nA.z, nA.w, nB.x, nB.y, nB.z, nB.w};
        float wa[8] = {w0A.x, w0A.y, w0A.z, w0A.w, w0B.x, w0B.y, w0B.z, w0B.w};
        float wb[8] = {w1A.x, w1A.y, w1A.z, w1A.w, w1B.x, w1B.y, w1B.z, w1B.w};

        #pragma unroll
        for (int r = 0; r < 8; ++r) {
            float sq   = fmaf(-2.0f, c[r], nr[r] + nxl);     // ||a||^2+||b||^2-2ab
            float dsq  = fmaxf(sq, 1e-12f);
            float d    = __builtin_amdgcn_sqrtf(dsq);        // v_sqrt_f32
            float s5d  = 2.2360679774997896f * d;            // sqrt(5)*d
            float poly = fmaf(1.6666666666666667f, dsq, 1.0f + s5d);
            float e    = __expf(-s5d);                       // v_exp_f32 path
            float k    = poly * e;
            acc0 = fmaf(wa[r], k, acc0);
            acc1 = fmaf(wb[r], k, acc1);
        }
    }

    // combine the two half-wave partials (column n held by lanes n and n+16)
    acc0 += __shfl_xor(acc0, 16, 32);
    acc1 += __shfl_xor(acc1, 16, 32);
    if (lane < 16) {
        ws[WS_PART0 + slice * N_X + j] = acc0;
        ws[WS_PART1 + slice * N_X + j] = acc1;
    }
}

// ---------------------------------------------------------------------------
// Kernel 4: deterministic slice reduction + treatment blend.
// ---------------------------------------------------------------------------
__global__ void combine(const float* __restrict__ w,
                        const float* __restrict__ ws,
                        float* __restrict__ out) {
    int j = blockIdx.x * blockDim.x + threadIdx.x;
    if (j >= N_X) return;
    const float* p0 = ws + WS_PART0;
    const float* p1 = ws + WS_PART1;
    float f0 = 0.0f, f1 = 0.0f;
    #pragma unroll
    for (int s = 0; s < N_SLICES; ++s) {
        f0 += p0[s * N_X + j];
        f1 += p1[s * N_X + j];
    }
    float wv = w[j];
    out[j] = wv * f0 + (1.0f - wv) * f1;
}

extern "C" void kernel_launch(void* const* d_in, const int* in_sizes, int n_in,
                              void* d_out, int out_size, void* d_ws, size_t ws_size,
                              hipStream_t stream) {
    const float* xobs = (const float*)d_in[0];
    const float* x    = (const float*)d_in[1];
    const float* w    = (const float*)d_in[2];
    const float* a0   = (const float*)d_in[3];
    const float* a1   = (const float*)d_in[4];
    const float* tfl  = (const float*)d_in[5];
    const float* rls  = (const float*)d_in[6];
    const int*   ro   = (const int*)d_in[7];
    const int*   rx   = (const int*)d_in[8];
    float* ws  = (float*)d_ws;
    float* out = (float*)d_out;

    prep_scalars<<<1, 32, 0, stream>>>(tfl, rls, ws);
    prep_rows<<<(N_OBS + N_X + 255) / 256, 256, 0, stream>>>(xobs, x, a0, a1, ro, ws);
    matern_wmma<<<(COL_TILES * N_SLICES) / 8, 256, 0, stream>>>(rx, ws);
    combine<<<(N_X + 255) / 256, 256, 0, stream>>>(w, ws, out);
}